// FNO1d_2688649527968
// MI455X (gfx1250) — compile-verified
//
#include <hip/hip_runtime.h>
#include <math.h>

// ---------------------------------------------------------------------------
// FNO1d for MI455X (gfx1250): truncated-DFT-as-GEMM, all big GEMMs on the
// fp32 WMMA pipe (V_WMMA_F32_16X16X4_F32), wave32, one 16-row tile per wave.
// Round 2: scalarized (SGPR) loop bounds via readfirstlane, pair-packed
// B-operand layouts for b64 fragment loads, prefetch on the HBM A-stream.
// ---------------------------------------------------------------------------

typedef __attribute__((ext_vector_type(2))) float v2f;
typedef __attribute__((ext_vector_type(8))) float v8f;

#define B_    64
#define W_    64
#define N_    8192
#define NPAD  8201          // n = 8192 + 9 (reference PAD)
#define NS    8208          // row stride, multiple of 16 (pad cols kept at 0)
#define M_    16
#define C32   32            // 16 real + 16 imag coefficients
#define H_    128           // projection hidden width
#define ROWS  4096          // B_*W_
#define KCH   12            // K-split chunks for forward DFT
#define TPC   171           // (NS/4)/KCH = 2052/12 K-steps per chunk

__device__ __forceinline__ float gelu_exact(float x) {
    return 0.5f * x * (1.0f + erff(x * 0.70710678118654752f));
}

__device__ __forceinline__ v8f wmma_f32(v2f a, v2f b, v8f c) {
    return __builtin_amdgcn_wmma_f32_16x16x4_f32(
        false, a, false, b, (short)0, c, false, false);
}

// ------------------------- basis tables ------------------------------------
// bfp : forward DFT basis, PAIR-PACKED: bfp[((j>>1)*32 + c)*2 + (j&1)]
//       c<16: cos(2*pi*j*c/n), c>=16: -sin(...)  (rows j>=NPAD forced to 0)
// ibp : inverse basis, PAIR-PACKED: ibp[((c>>1)*NS + j)*2 + (c&1)],
//       scaled by 1/n (k==0) or 2/n (k>0).
__global__ void k_basis(float* __restrict__ bfp, float* __restrict__ ibp) {
    int idx = blockIdx.x * blockDim.x + threadIdx.x;
    if (idx >= NS * C32) return;
    int jj = idx >> 5, c = idx & 31, k = c & 15;
    float cv = 0.0f, iv = 0.0f;
    if (jj < NPAD) {
        int m = (jj * k) % NPAD;                 // exact integer phase reduction
        float ph = 6.28318530717958647692f * (float)m / (float)NPAD;
        float s, co;
        sincosf(ph, &s, &co);
        float alpha = (k == 0) ? (1.0f / (float)NPAD) : (2.0f / (float)NPAD);
        cv = (c < 16) ? co : -s;
        iv = (c < 16) ? alpha * co : -alpha * s;
    }
    bfp[(((jj >> 1) * C32) + c) * 2 + (jj & 1)]       = cv;
    ibp[(((size_t)(c >> 1) * NS) + jj) * 2 + (c & 1)] = iv;
}

// pair-pack proj_w1: w1p[((k>>1)*H + h)*2 + (k&1)] = w1[k*H + h]
__global__ void k_pack_w1(const float* __restrict__ w1, float* __restrict__ w1p) {
    int idx = blockIdx.x * blockDim.x + threadIdx.x;
    if (idx >= W_ * H_) return;
    int k = idx / H_, h = idx % H_;
    w1p[(((k >> 1) * H_) + h) * 2 + (k & 1)] = w1[idx];
}

// ------------------------- lift: v = x @ lift_w + lift_b -------------------
__global__ void k_lift(const float* __restrict__ x, const float* __restrict__ lw,
                       const float* __restrict__ lb, float* __restrict__ v) {
    int idx = blockIdx.x * blockDim.x + threadIdx.x;
    if (idx >= ROWS * NS) return;
    int j = idx % NS, r = idx / NS;
    int w = r & 63, b = r >> 6;
    float val = 0.0f;
    if (j < N_) {
        const float* xp = x + ((size_t)b * N_ + j) * 2;
        val = xp[0] * lw[w] + xp[1] * lw[64 + w] + lb[w];
    }
    v[idx] = val;
}

__global__ void k_zero(float* __restrict__ p, int nn) {
    int i = blockIdx.x * blockDim.x + threadIdx.x;
    if (i < nn) p[i] = 0.0f;
}

// ------------------------- forward truncated DFT (WMMA GEMM) ---------------
// F(4096 x 32) += V(4096 x NS) @ bf(NS x 32). One wave = 16 rows x 32 cols,
// K split into KCH chunks, fp32 global atomics accumulate partials.
__global__ void k_fft(const float* __restrict__ v, const float* __restrict__ bfp,
                      float* __restrict__ F) {
    const int wid  = __builtin_amdgcn_readfirstlane(
                        (blockIdx.x * blockDim.x + threadIdx.x) >> 5);
    const int lane = threadIdx.x & 31;
    const int half = lane >> 4, lm = lane & 15;
    const int rt = wid / KCH, ch = wid % KCH;
    const int r0 = rt << 4;
    const float* arow = v + (size_t)(r0 + lm) * NS;
    v8f c0 = {}; v8f c1 = {};
    const int t0 = ch * TPC, t1 = t0 + TPC;       // SGPR loop bounds
#pragma unroll 2
    for (int t = t0; t < t1; ++t) {
        const int k = (t << 2) + (half << 1);     // K index for this half
        const int p = (t << 1) + half;            // pair index (k >> 1)
        if ((t & 15) == 0)                        // uniform scalar branch
            __builtin_prefetch(arow + (t << 2) + 256, 0, 1);
        v2f a  = *(const v2f*)(arow + k);
        v2f b0 = *(const v2f*)(bfp + (p * C32 + lm) * 2);
        v2f b1 = *(const v2f*)(bfp + (p * C32 + 16 + lm) * 2);
        c0 = wmma_f32(a, b0, c0);
        c1 = wmma_f32(a, b1, c1);
    }
#pragma unroll
    for (int m = 0; m < 8; ++m) {
        const int row = r0 + m + (half << 3);
        atomicAdd(&F[row * C32 + lm],      c0[m]);
        atomicAdd(&F[row * C32 + 16 + lm], c1[m]);
    }
}

// ------------------------- per-mode complex 64x64 mix (tiny, VALU) ---------
__global__ void k_mix(const float* __restrict__ F, const float* __restrict__ kr,
                      const float* __restrict__ ki, float* __restrict__ P) {
    int idx = blockIdx.x * blockDim.x + threadIdx.x;
    if (idx >= B_ * W_ * M_) return;
    int k = idx & 15, o = (idx >> 4) & 63, b = idx >> 10;
    float pr = 0.0f, pi = 0.0f;
#pragma unroll 4
    for (int i = 0; i < 64; ++i) {
        float fr  = F[(b * 64 + i) * C32 + k];
        float fi  = F[(b * 64 + i) * C32 + 16 + k];
        float krv = kr[((i * 64) + o) * 16 + k];
        float kiv = ki[((i * 64) + o) * 16 + k];
        pr += fr * krv - fi * kiv;
        pi += fr * kiv + fi * krv;
    }
    P[(b * 64 + o) * C32 + k]      = pr;
    P[(b * 64 + o) * C32 + 16 + k] = pi;
}

// ------------------- fused irfft + channel-mix + bias + GELU ---------------
// vout[b,o,j] = gelu( sum_i cw[o,i] v[b,i,j] + sum_c P[b,o,c] ibas[c,j] + cb[o] )
__global__ void k_inv(const float* __restrict__ vin, const float* __restrict__ P,
                      const float* __restrict__ ibp, const float* __restrict__ cw,
                      const float* __restrict__ cb, float* __restrict__ vout) {
    const int wid  = __builtin_amdgcn_readfirstlane(
                        (blockIdx.x * blockDim.x + threadIdx.x) >> 5);
    const int lane = threadIdx.x & 31;
    const int half = lane >> 4, lm = lane & 15;
    const int rt = wid / 513, jt = wid % 513;
    const int r0 = rt << 4, b = r0 >> 6, o0 = r0 & 63, j0 = jt << 4;
    const int j = j0 + lm;
    v8f c = {};
#pragma unroll
    for (int t = 0; t < 16; ++t) {                 // linear term, K = 64 (i)
        const int k = (t << 2) + (half << 1);
        v2f a = *(const v2f*)(cw + (o0 + lm) * 64 + k);   // contiguous -> b64
        v2f bb; bb.x = vin[(size_t)(b * 64 + k) * NS + j];
                bb.y = vin[(size_t)(b * 64 + k + 1) * NS + j];
        c = wmma_f32(a, bb, c);
    }
#pragma unroll
    for (int t = 0; t < 8; ++t) {                  // spectral term, K = 32 (c)
        const int k = (t << 2) + (half << 1);
        const int p = (t << 1) + half;
        v2f a  = *(const v2f*)(P + (r0 + lm) * C32 + k);  // contiguous -> b64
        v2f bb = *(const v2f*)(ibp + ((size_t)p * NS + j) * 2);
        c = wmma_f32(a, bb, c);
    }
#pragma unroll
    for (int m = 0; m < 8; ++m) {
        const int row = r0 + m + (half << 3);
        const int o   = o0 + m + (half << 3);
        float g = gelu_exact(c[m] + cb[o]);
        vout[(size_t)row * NS + j] = (j < NPAD) ? g : 0.0f;   // keep stride pad 0
    }
}

// ------------------------- projection MLP (WMMA) ---------------------------
// out[b,j] = sum_h gelu( sum_w v[b,w,j] W1[w,h] + b1[h] ) * W2[h] + b2
__global__ void k_proj(const float* __restrict__ v, const float* __restrict__ w1p,
                       const float* __restrict__ b1, const float* __restrict__ w2,
                       const float* __restrict__ b2, float* __restrict__ out) {
    const int wid  = __builtin_amdgcn_readfirstlane(
                        (blockIdx.x * blockDim.x + threadIdx.x) >> 5);
    const int lane = threadIdx.x & 31;
    const int half = lane >> 4, lm = lane & 15;
    const int b = wid >> 9, pt = wid & 511, j0 = pt << 4;
    const int j = j0 + lm;
    float acc[8] = {0.f, 0.f, 0.f, 0.f, 0.f, 0.f, 0.f, 0.f};
    for (int nt = 0; nt < 8; ++nt) {
        v8f c = {};
        const int hcol = (nt << 4) + lm;
#pragma unroll
        for (int t = 0; t < 16; ++t) {
            const int k = (t << 2) + (half << 1);
            const int p = (t << 1) + half;
            v2f a;  a.x = v[(size_t)(b * 64 + k) * NS + j];
                    a.y = v[(size_t)(b * 64 + k + 1) * NS + j];
            v2f bb = *(const v2f*)(w1p + (p * H_ + hcol) * 2);
            c = wmma_f32(a, bb, c);
        }
        const float wh = w2[hcol];
        const float bh = b1[hcol];
#pragma unroll
        for (int m = 0; m < 8; ++m) acc[m] += gelu_exact(c[m] + bh) * wh;
    }
#pragma unroll
    for (int m = 0; m < 8; ++m) {
        for (int off = 8; off > 0; off >>= 1)      // reduce over 16-lane group
            acc[m] += __shfl_xor(acc[m], off, 32);
    }
    if (lm == 0) {
#pragma unroll
        for (int m = 0; m < 8; ++m) {
            const int jj = j0 + m + (half << 3);
            out[(size_t)b * N_ + jj] = acc[m] + b2[0];
        }
    }
}

// ---------------------------------------------------------------------------
extern "C" void kernel_launch(void* const* d_in, const int* in_sizes, int n_in,
                              void* d_out, int out_size, void* d_ws, size_t ws_size,
                              hipStream_t stream) {
    const float* x      = (const float*)d_in[0];
    const float* lift_w = (const float*)d_in[1];
    const float* lift_b = (const float*)d_in[2];
    const float* kern_r = (const float*)d_in[3];
    const float* kern_i = (const float*)d_in[4];
    const float* conv_w = (const float*)d_in[5];
    const float* conv_b = (const float*)d_in[6];
    const float* pw1    = (const float*)d_in[7];
    const float* pb1    = (const float*)d_in[8];
    const float* pw2    = (const float*)d_in[9];
    const float* pb2    = (const float*)d_in[10];

    float* ws   = (float*)d_ws;
    float* v0   = ws;                                  // (4096, NS)
    float* v1   = v0   + (size_t)ROWS * NS;            // (4096, NS)
    float* bfp  = v1   + (size_t)ROWS * NS;            // (NS/2, 32, 2) pair-packed
    float* ibp  = bfp  + (size_t)NS * C32;             // (16, NS, 2) pair-packed
    float* F    = ibp  + (size_t)C32 * NS;             // (4096, 32)
    float* P    = F    + (size_t)ROWS * C32;           // (4096, 32)
    float* w1p  = P    + (size_t)ROWS * C32;           // (32, 128, 2) pair-packed

    k_basis<<<(NS * C32 + 255) / 256, 256, 0, stream>>>(bfp, ibp);
    k_pack_w1<<<(W_ * H_ + 255) / 256, 256, 0, stream>>>(pw1, w1p);
    k_lift<<<(ROWS * NS + 255) / 256, 256, 0, stream>>>(x, lift_w, lift_b, v0);

    float* vcur = v0;
    float* vnext = v1;
    for (int l = 0; l < 5; ++l) {
        k_zero<<<(ROWS * C32) / 256, 256, 0, stream>>>(F, ROWS * C32);
        k_fft<<<(256 * KCH) / 8, 256, 0, stream>>>(vcur, bfp, F);         // 3072 waves
        k_mix<<<(B_ * W_ * M_) / 256, 256, 0, stream>>>(
            F, kern_r + (size_t)l * 65536, kern_i + (size_t)l * 65536, P);
        k_inv<<<(256 * 513) / 8, 256, 0, stream>>>(                       // 131328 waves
            vcur, P, ibp, conv_w + (size_t)l * 4096, conv_b + (size_t)l * 64, vnext);
        float* tmp = vcur; vcur = vnext; vnext = tmp;
    }
    k_proj<<<(B_ * 512) / 8, 256, 0, stream>>>(vcur, pw1, pb1, pw2, pb2, (float*)d_out);
}